// GraphEncoder_3753801416793
// MI455X (gfx1250) — compile-verified
//
#include <hip/hip_runtime.h>
#include <math.h>

typedef __attribute__((ext_vector_type(2))) float v2f;
typedef __attribute__((ext_vector_type(8))) float v8f;

#define IN_DIM   128
#define OUT_DIM  32
#define LEAKY    0.01f

// ---------------------------------------------------------------------------
// Monotone mapping float -> uint so unsigned atomicMax == float max
// (handles negatives from leaky_relu correctly). Init key = 0 == "-inf".
// ---------------------------------------------------------------------------
__device__ __forceinline__ unsigned fkey(float f) {
  unsigned u = __float_as_uint(f);
  return (u & 0x80000000u) ? ~u : (u | 0x80000000u);
}
__device__ __forceinline__ float funkey(unsigned k) {
  unsigned u = (k & 0x80000000u) ? (k ^ 0x80000000u) : ~k;
  return __uint_as_float(u);
}

// ---------------------------------------------------------------------------
// Kernel 1: z = h @ W^T via V_WMMA_F32_16X16X4_F32 (fp32 in, fp32 acc).
// One wave computes a 16x32 tile of z (two 16x16 WMMA tiles), K in steps of 4.
// Branchless: A row is clamped for tail tiles (duplicate compute, guarded
// stores), so EXEC stays full and loads issue in clauses.
//
// Per-lane layouts (CDNA5 ISA 7.12.2):
//   A 16x4 f32 : lanes 0-15 -> M=lane, VGPR0=K0,VGPR1=K1 ; lanes 16-31 -> K2,K3
//   B 4x16 f32 : lanes 0-15 -> N=lane, VGPR0=K0,VGPR1=K1 ; lanes 16-31 -> K2,K3
//   C/D 16x16  : VGPR r -> M=r (lanes 0-15) / M=8+r (lanes 16-31), N=lane&15
// ---------------------------------------------------------------------------
__global__ void __launch_bounds__(256)
gat_gemm_wmma(const float* __restrict__ h, const float* __restrict__ W,
              float* __restrict__ z, int n_nodes) {
  const int lane   = threadIdx.x & 31;
  const int wave   = threadIdx.x >> 5;
  const int tile_m = blockIdx.x * 8 + wave;
  const int row0   = tile_m * 16;
  if (row0 >= n_nodes) return;

  const int l15 = lane & 15;
  const int kh  = lane >> 4;               // which K half-pair this lane holds
  int arow = row0 + l15;
  arow = (arow < n_nodes) ? arow : (n_nodes - 1);   // clamp: keep EXEC full

  const float* pa  = h + (size_t)arow * IN_DIM + 2 * kh;
  const float* pb0 = W + (size_t)l15 * IN_DIM + 2 * kh;   // out cols 0..15
  const float* pb1 = pb0 + (size_t)16 * IN_DIM;           // out cols 16..31

  v8f c0 = {};
  v8f c1 = {};
  #pragma unroll 8
  for (int k = 0; k < IN_DIM; k += 4) {
    const v2f a  = *(const v2f*)pa;
    const v2f b0 = *(const v2f*)pb0;
    const v2f b1 = *(const v2f*)pb1;
    pa += 4; pb0 += 4; pb1 += 4;
    c0 = __builtin_amdgcn_wmma_f32_16x16x4_f32(false, a, false, b0,
                                               (short)0, c0, false, false);
    c1 = __builtin_amdgcn_wmma_f32_16x16x4_f32(false, a, false, b1,
                                               (short)0, c1, false, false);
  }

  #pragma unroll
  for (int r = 0; r < 8; ++r) {
    const int row = row0 + r + 8 * kh;
    if (row < n_nodes) {
      z[(size_t)row * OUT_DIM + l15]      = c0[r];
      z[(size_t)row * OUT_DIM + 16 + l15] = c1[r];
    }
  }
}

// ---------------------------------------------------------------------------
// Kernel 2: e_s[i] = z[i] . attn_w[0:32],  e_d[i] = z[i] . attn_w[32:64]
// Lane-per-column (coalesced B32 loads) + wave32 xor-shuffle reduction.
// ---------------------------------------------------------------------------
__global__ void __launch_bounds__(256)
gat_attn_proj(const float* __restrict__ z, const float* __restrict__ attn_w,
              float* __restrict__ e_s, float* __restrict__ e_d, int n_nodes) {
  const int lane = threadIdx.x & 31;
  const int node = (blockIdx.x * blockDim.x + threadIdx.x) >> 5;
  if (node >= n_nodes) return;
  const float zv = z[(size_t)node * OUT_DIM + lane];
  float ps = zv * attn_w[lane];
  float pd = zv * attn_w[OUT_DIM + lane];
  #pragma unroll
  for (int off = 16; off > 0; off >>= 1) {
    ps += __shfl_xor(ps, off, 32);
    pd += __shfl_xor(pd, off, 32);
  }
  if (lane == 0) { e_s[node] = ps; e_d[node] = pd; }
}

// ---------------------------------------------------------------------------
// Kernel 3: init out = 0, denom = 0, max-keys = 0 ("-inf")
// ---------------------------------------------------------------------------
__global__ void __launch_bounds__(256)
gat_init(float* __restrict__ out, float* __restrict__ denom,
         unsigned* __restrict__ mkey, int n_nodes) {
  const int tid = blockIdx.x * blockDim.x + threadIdx.x;
  if (tid < n_nodes * OUT_DIM) out[tid] = 0.0f;
  if (tid < n_nodes) { denom[tid] = 0.0f; mkey[tid] = 0u; }
}

// ---------------------------------------------------------------------------
// Kernel 4: e = leaky_relu(e_s[src] + e_d[dst]); segment max via u32 atomicMax
// ---------------------------------------------------------------------------
__global__ void __launch_bounds__(256)
gat_edge_max(const float* __restrict__ e_s, const float* __restrict__ e_d,
             const int* __restrict__ src, const int* __restrict__ dst,
             float* __restrict__ ebuf, unsigned* __restrict__ mkey,
             int n_edges) {
  const int e = blockIdx.x * blockDim.x + threadIdx.x;
  if (e >= n_edges) return;
  const float x = e_s[src[e]] + e_d[dst[e]];
  const float v = (x > 0.0f) ? x : LEAKY * x;
  ebuf[e] = v;
  atomicMax(&mkey[dst[e]], fkey(v));
}

// ---------------------------------------------------------------------------
// Kernel 5: ex = exp(e - m[dst]); denom[dst] += ex  (f32 atomic add)
// ---------------------------------------------------------------------------
__global__ void __launch_bounds__(256)
gat_edge_exp(const int* __restrict__ dst, const unsigned* __restrict__ mkey,
             float* __restrict__ ebuf, float* __restrict__ denom, int n_edges) {
  const int e = blockIdx.x * blockDim.x + threadIdx.x;
  if (e >= n_edges) return;
  const int d = dst[e];
  const float ex = __expf(ebuf[e] - funkey(mkey[d]));
  ebuf[e] = ex;
  atomicAdd(&denom[d], ex);
}

// ---------------------------------------------------------------------------
// Kernel 6: out[dst] += (ex/denom[dst]) * z[src]
// One wave per edge: lane = output column -> coalesced z gather + f32 atomics.
// ---------------------------------------------------------------------------
__global__ void __launch_bounds__(256)
gat_edge_scatter(const int* __restrict__ src, const int* __restrict__ dst,
                 const float* __restrict__ z, const float* __restrict__ ebuf,
                 const float* __restrict__ denom, float* __restrict__ out,
                 int n_edges) {
  const int tid  = blockIdx.x * blockDim.x + threadIdx.x;
  const int e    = tid >> 5;
  const int lane = tid & 31;
  if (e >= n_edges) return;
  const int s = src[e];
  const int d = dst[e];
  float dn = denom[d];
  dn = (dn == 0.0f) ? 1.0f : dn;
  const float alpha = ebuf[e] / dn;
  const float zv = z[(size_t)s * OUT_DIM + lane];
  atomicAdd(&out[(size_t)d * OUT_DIM + lane], alpha * zv);
}

// ---------------------------------------------------------------------------
extern "C" void kernel_launch(void* const* d_in, const int* in_sizes, int n_in,
                              void* d_out, int out_size, void* d_ws, size_t ws_size,
                              hipStream_t stream) {
  const float* h      = (const float*)d_in[0];
  const float* W      = (const float*)d_in[1];
  const float* attn_w = (const float*)d_in[2];
  const int*   src    = (const int*)d_in[3];
  const int*   dst    = (const int*)d_in[4];

  const int n_nodes = in_sizes[0] / IN_DIM;
  const int n_edges = in_sizes[3];

  // Workspace partition
  float*    z     = (float*)d_ws;                       // n_nodes*32
  float*    e_s   = z + (size_t)n_nodes * OUT_DIM;      // n_nodes
  float*    e_d   = e_s + n_nodes;                      // n_nodes
  unsigned* mkey  = (unsigned*)(e_d + n_nodes);         // n_nodes
  float*    denom = (float*)(mkey + n_nodes);           // n_nodes
  float*    ebuf  = denom + n_nodes;                    // n_edges
  float*    out   = (float*)d_out;

  const int row_tiles   = (n_nodes + 15) / 16;
  const int gemm_blocks = (row_tiles + 7) / 8;
  gat_gemm_wmma<<<gemm_blocks, 256, 0, stream>>>(h, W, z, n_nodes);

  const int nthreads_node32 = n_nodes * OUT_DIM;
  gat_attn_proj<<<(nthreads_node32 + 255) / 256, 256, 0, stream>>>(
      z, attn_w, e_s, e_d, n_nodes);

  gat_init<<<(nthreads_node32 + 255) / 256, 256, 0, stream>>>(
      out, denom, mkey, n_nodes);

  gat_edge_max<<<(n_edges + 255) / 256, 256, 0, stream>>>(
      e_s, e_d, src, dst, ebuf, mkey, n_edges);

  gat_edge_exp<<<(n_edges + 255) / 256, 256, 0, stream>>>(
      dst, mkey, ebuf, denom, n_edges);

  const long long scat_threads = (long long)n_edges * 32;
  gat_edge_scatter<<<(int)((scat_threads + 255) / 256), 256, 0, stream>>>(
      src, dst, z, ebuf, denom, out, n_edges);
}